// GlobalAttentionGNN_64991445123836
// MI455X (gfx1250) — compile-verified
//
#include <hip/hip_runtime.h>

// GlobalAttentionGNN for MI455X (gfx1250, wave32).
// Pipeline: zero -> edge scatter-add (L2 atomics) -> WMMA f32 GEMM+ReLU+scores
//           -> segment max -> exp/segment-sum -> weighted pooling scatter.
// Memory-bound workload (~1 GB of mostly L2-resident traffic); the 64x64 GEMM
// uses V_WMMA_F32_16X16X4_F32 so full fp32 precision is kept while the matrix
// pipe stays far off the critical path. W is staged TRANSPOSED in LDS (pitch
// 68) so every B fragment is a single conflict-free ds_load_b64, matching the
// A-fragment path and eliminating VGPR repack movs.

#define GN_NODES  100000
#define GN_EDGES  1600000
#define GN_GRAPHS 256
#define GN_F      64

typedef __attribute__((ext_vector_type(2))) float v2f;
typedef __attribute__((ext_vector_type(8))) float v8f;

// ---------------- utility: order-preserving float<->uint for atomic max ----
__device__ __forceinline__ unsigned fmap_u(float f) {
    unsigned u = __float_as_uint(f);
    return (u & 0x80000000u) ? ~u : (u | 0x80000000u);
}
__device__ __forceinline__ float funmap_u(unsigned u) {
    unsigned v = (u & 0x80000000u) ? (u & 0x7FFFFFFFu) : ~u;
    return __uint_as_float(v);
}

// ---------------- zero kernels --------------------------------------------
__global__ void gnn_zero_f4(float4* __restrict__ p, int n4) {
    int i = blockIdx.x * blockDim.x + threadIdx.x;
    if (i < n4) p[i] = make_float4(0.f, 0.f, 0.f, 0.f);
}
__global__ void gnn_zero_seg(unsigned* __restrict__ smax, float* __restrict__ denom) {
    int i = threadIdx.x;
    if (i < GN_GRAPHS) { smax[i] = 0u; denom[i] = 0.f; }
}

// ---------------- edge scatter: agg[dst] += x[src] -------------------------
// 16 threads per edge, each moves one float4 (coalesced 256B per edge).
__global__ __launch_bounds__(256) void gnn_edge_scatter(
    const float* __restrict__ x, const long long* __restrict__ ei,
    float* __restrict__ agg) {
    int t = blockIdx.x * blockDim.x + threadIdx.x;   // < GN_EDGES*16
    int e  = t >> 4;
    int fc = t & 15;
    int src = (int)ei[e];
    int dst = (int)ei[GN_EDGES + e];
    float4 v = ((const float4*)x)[src * 16 + fc];
    float* d = agg + (size_t)dst * GN_F + fc * 4;
    atomicAdd(d + 0, v.x);
    atomicAdd(d + 1, v.y);
    atomicAdd(d + 2, v.z);
    atomicAdd(d + 3, v.w);
}

// ---------------- fused GEMM (WMMA f32) + bias + ReLU + attention scores ---
// Block = 256 threads = 8 waves; each wave owns a 16-node x 64-feature tile.
// agg buffer is overwritten in place with h = relu((x+agg)W + b).
#define SROW 68   // LDS row pitch in floats (68 mod 64 = 4 -> conflict-free b64 loads)
__global__ __launch_bounds__(256) void gnn_gemm_relu_score(
    const float* __restrict__ x, float* __restrict__ sumbuf,
    const float* __restrict__ Wg, const float* __restrict__ bg,
    const float* __restrict__ Wa, const float* __restrict__ ba,
    float* __restrict__ scores, int n_nodes) {
    __shared__ float sWt[GN_F * SROW];     // 17 KB: W transposed, sWt[n*SROW + k]
    __shared__ float sT[128 * SROW];       // 34 KB: input tile -> h tile
    __shared__ float sAtt[GN_F];

    const int tid = threadIdx.x;
    // stage W_gnn transposed: sWt[n][k] = Wg[k][n]
    for (int i = tid; i < GN_F * (GN_F / 4); i += 256) {
        int k  = i >> 4;          // row of W
        int c4 = i & 15;          // group of 4 columns
        float4 w = ((const float4*)Wg)[i];
        sWt[(c4 * 4 + 0) * SROW + k] = w.x;
        sWt[(c4 * 4 + 1) * SROW + k] = w.y;
        sWt[(c4 * 4 + 2) * SROW + k] = w.z;
        sWt[(c4 * 4 + 3) * SROW + k] = w.w;
    }
    if (tid < GN_F) sAtt[tid] = Wa[tid];

    const int base = blockIdx.x * 128;
    // stage (x + agg) tile, zero-padded past n_nodes
    for (int i = tid; i < 128 * 16; i += 256) {
        int r  = i >> 4;
        int c4 = i & 15;
        float4 v = make_float4(0.f, 0.f, 0.f, 0.f);
        int row = base + r;
        if (row < n_nodes) {
            float4 a = ((const float4*)x)[row * 16 + c4];
            float4 g = ((const float4*)sumbuf)[row * 16 + c4];
            v = make_float4(a.x + g.x, a.y + g.y, a.z + g.z, a.w + g.w);
        }
        ((float4*)(sT + r * SROW))[c4] = v;
    }
    __syncthreads();

    const int wave = tid >> 5;
    const int lane = tid & 31;
    const int m  = lane & 15;   // M (A) / N (B,C) position
    const int h2 = lane >> 4;   // lane-half selects K pair
    const int rowBase = wave * 16;
    const float* aRow = sT + (rowBase + m) * SROW;
    const float* bCol0 = sWt + ( 0 + m) * SROW;
    const float* bCol1 = sWt + (16 + m) * SROW;
    const float* bCol2 = sWt + (32 + m) * SROW;
    const float* bCol3 = sWt + (48 + m) * SROW;

    v8f c0 = {}, c1 = {}, c2 = {}, c3 = {};
    for (int kk = 0; kk < 16; ++kk) {
        const int k0 = kk * 4 + 2 * h2;           // even -> 8B aligned b64 loads
        v2f a  = *(const v2f*)(aRow  + k0);
        v2f b0 = *(const v2f*)(bCol0 + k0);
        v2f b1 = *(const v2f*)(bCol1 + k0);
        v2f b2 = *(const v2f*)(bCol2 + k0);
        v2f b3 = *(const v2f*)(bCol3 + k0);
        c0 = __builtin_amdgcn_wmma_f32_16x16x4_f32(false, a, false, b0, (short)0, c0, false, false);
        c1 = __builtin_amdgcn_wmma_f32_16x16x4_f32(false, a, false, b1, (short)0, c1, false, false);
        c2 = __builtin_amdgcn_wmma_f32_16x16x4_f32(false, a, false, b2, (short)0, c2, false, false);
        c3 = __builtin_amdgcn_wmma_f32_16x16x4_f32(false, a, false, b3, (short)0, c3, false, false);
    }

    // bias + ReLU; write h to LDS (wave-private rows) and to global (in place)
#pragma unroll
    for (int r = 0; r < 8; ++r) {
        const int rowIn = rowBase + r + 8 * h2;   // C layout: VGPR r = rows r, r+8
        const int grow  = base + rowIn;
        float hv;
        hv = fmaxf(c0[r] + bg[ 0 + m], 0.f); sT[rowIn * SROW +  0 + m] = hv;
        if (grow < n_nodes) sumbuf[(size_t)grow * GN_F +  0 + m] = hv;
        hv = fmaxf(c1[r] + bg[16 + m], 0.f); sT[rowIn * SROW + 16 + m] = hv;
        if (grow < n_nodes) sumbuf[(size_t)grow * GN_F + 16 + m] = hv;
        hv = fmaxf(c2[r] + bg[32 + m], 0.f); sT[rowIn * SROW + 32 + m] = hv;
        if (grow < n_nodes) sumbuf[(size_t)grow * GN_F + 32 + m] = hv;
        hv = fmaxf(c3[r] + bg[48 + m], 0.f); sT[rowIn * SROW + 48 + m] = hv;
        if (grow < n_nodes) sumbuf[(size_t)grow * GN_F + 48 + m] = hv;
    }
    __syncthreads();

    // attention scores: 2 threads/node, 32 features each, shuffle combine
    const int rnode = tid >> 1;
    const int half  = tid & 1;
    float s = 0.f;
#pragma unroll 8
    for (int j = 0; j < 32; ++j) {
        int col = half * 32 + j;
        s += sT[rnode * SROW + col] * sAtt[col];
    }
    s += __shfl_xor(s, 1);
    const int grow = base + rnode;
    if (half == 0 && grow < n_nodes) scores[grow] = s + ba[0];
}

// ---------------- segment max (atomic on order-preserving uint) ------------
__global__ void gnn_seg_max(const float* __restrict__ scores,
                            const long long* __restrict__ batch,
                            unsigned* __restrict__ smax) {
    int i = blockIdx.x * blockDim.x + threadIdx.x;
    if (i >= GN_NODES) return;
    atomicMax(&smax[(int)batch[i]], fmap_u(scores[i]));
}

// ---------------- e = exp(score - max); denom += e (scores -> e in place) --
__global__ void gnn_exp_sum(float* __restrict__ scores,
                            const long long* __restrict__ batch,
                            const unsigned* __restrict__ smax,
                            float* __restrict__ denom) {
    int i = blockIdx.x * blockDim.x + threadIdx.x;
    if (i >= GN_NODES) return;
    int b = (int)batch[i];
    float e = expf(scores[i] - funmap_u(smax[b]));
    scores[i] = e;
    atomicAdd(&denom[b], e);
}

// ---------------- weighted pooling: out[b] += (e/denom) * h ----------------
__global__ __launch_bounds__(256) void gnn_pool(
    const float* __restrict__ h, const float* __restrict__ e,
    const long long* __restrict__ batch, const float* __restrict__ denom,
    float* __restrict__ out) {
    int t = blockIdx.x * blockDim.x + threadIdx.x;
    if (t >= GN_NODES * 16) return;
    int i  = t >> 4;
    int fc = t & 15;
    int b  = (int)batch[i];
    float w = e[i] / denom[b];
    float4 v = ((const float4*)h)[i * 16 + fc];
    float* o = out + b * GN_F + fc * 4;
    atomicAdd(o + 0, w * v.x);
    atomicAdd(o + 1, w * v.y);
    atomicAdd(o + 2, w * v.z);
    atomicAdd(o + 3, w * v.w);
}

extern "C" void kernel_launch(void* const* d_in, const int* in_sizes, int n_in,
                              void* d_out, int out_size, void* d_ws, size_t ws_size,
                              hipStream_t stream) {
    const float*      x     = (const float*)d_in[0];
    const long long*  ei    = (const long long*)d_in[1];   // int64 (2, E)
    const long long*  batch = (const long long*)d_in[2];   // int64 (N)
    const float*      Wg    = (const float*)d_in[3];
    const float*      bg    = (const float*)d_in[4];
    const float*      Wa    = (const float*)d_in[5];
    const float*      ba    = (const float*)d_in[6];
    float*            out   = (float*)d_out;

    // workspace layout (~26.1 MB): agg (becomes h in place) | scores (becomes e) | smax | denom
    float*    agg    = (float*)d_ws;
    float*    scores = agg + (size_t)GN_NODES * GN_F;
    unsigned* smax   = (unsigned*)(scores + GN_NODES);
    float*    denom  = (float*)(smax + GN_GRAPHS);

    (void)in_sizes; (void)n_in; (void)out_size; (void)ws_size;

    const int aggN4 = GN_NODES * GN_F / 4;
    const int outN4 = GN_GRAPHS * GN_F / 4;
    gnn_zero_f4<<<(aggN4 + 255) / 256, 256, 0, stream>>>((float4*)agg, aggN4);
    gnn_zero_f4<<<(outN4 + 255) / 256, 256, 0, stream>>>((float4*)out, outN4);
    gnn_zero_seg<<<1, 256, 0, stream>>>(smax, denom);

    gnn_edge_scatter<<<GN_EDGES * 16 / 256, 256, 0, stream>>>(x, ei, agg);

    gnn_gemm_relu_score<<<(GN_NODES + 127) / 128, 256, 0, stream>>>(
        x, agg, Wg, bg, Wa, ba, scores, GN_NODES);

    gnn_seg_max<<<(GN_NODES + 255) / 256, 256, 0, stream>>>(scores, batch, smax);
    gnn_exp_sum<<<(GN_NODES + 255) / 256, 256, 0, stream>>>(scores, batch, smax, denom);
    gnn_pool<<<(GN_NODES * 16 + 255) / 256, 256, 0, stream>>>(agg, scores, batch, denom, out);
}